// GCN_40157944218266
// MI455X (gfx1250) — compile-verified
//
#include <hip/hip_runtime.h>

// Fused 3-layer GCN for MI455X (gfx1250): one workgroup per graph,
// all intermediates in LDS, GEMMs via V_WMMA_F32_16X16X4_F32 (exact f32),
// edge aggregation via ds_add_f32 LDS atomics.

typedef __attribute__((ext_vector_type(2))) float v2f;
typedef __attribute__((ext_vector_type(8))) float v8f;

#define B_GRAPHS 4096
#define NPG      148     // nodes per graph
#define EPG      592     // edges per graph
#define NPAD     160     // nodes padded to multiple of 16 for WMMA tiling
#define HSTRIDE  32      // row stride (floats) of sH / sAcc
#define NTHREADS 256     // 8 wave32s

// One layer "column pass": computes cw (<=32) output columns [c0, c0+cw).
//   sX   : layer input,  row-major [NPAD x FI]
//   sH   : h = xW,       row-major [NPAD x 32]
//   sAcc : aggregated,   row-major [NPAD x 32]
template <int FI, int FO, int CW, bool LAST>
__device__ __forceinline__ void layer_pass(
    const float* __restrict__ W, const float* __restrict__ bias,
    float* sX, float* sH, float* sAcc,
    const unsigned* sEdge, const float* sDis,
    float* __restrict__ outg, int c0, int tid)
{
    const int lane   = tid & 31;
    const int wave   = tid >> 5;
    const int nwaves = NTHREADS / 32;
    const int mh     = lane & 15;      // M (or N) index within 16-half
    const int hi     = lane >> 4;      // 0 or 1: which half of the wave

    // ---- GEMM: sH[:, ct*16 + 0..15] = sX @ W[:, c0 + ct*16 + 0..15] ----
    const int ntile = NPAD / 16;       // 10 node tiles
    const int ctile = CW / 16;         // 1 or 2 column tiles
    for (int job = wave; job < ntile * ctile; job += nwaves) {
        const int t   = job / ctile;
        const int ct  = job % ctile;
        const int m0  = t * 16;
        const int col = c0 + ct * 16 + mh;     // global output column
        v8f c = {0.f, 0.f, 0.f, 0.f, 0.f, 0.f, 0.f, 0.f};
#pragma unroll
        for (int k0 = 0; k0 < FI; k0 += 4) {
            const int kb = k0 + hi * 2;        // ISA A/B layout: lanes16-31 hold K+2,K+3
            const int m  = m0 + mh;
            v2f a, b;
            a.x = sX[m * FI + kb];
            a.y = sX[m * FI + kb + 1];
            b.x = W[kb * FO + col];
            b.y = W[(kb + 1) * FO + col];
            c = __builtin_amdgcn_wmma_f32_16x16x4_f32(
                    false, a, false, b, (short)0, c, false, false);
        }
#pragma unroll
        for (int r = 0; r < 8; ++r) {
            const int m = m0 + r + hi * 8;     // C/D layout: VGPR r -> M=r / M=r+8
            sH[m * HSTRIDE + ct * 16 + mh] = c[r];
        }
    }
    __syncthreads();

    // ---- acc = self-loop term + bias ----
    for (int t = tid; t < NPG * CW; t += NTHREADS) {
        const int i = t / CW;
        const int f = t % CW;
        const float di = sDis[i];
        sAcc[i * HSTRIDE + f] = sH[i * HSTRIDE + f] * di * di + bias[c0 + f];
    }
    __syncthreads();

    // ---- edge scatter: acc[d] += h[s] * dis[s] * dis[d]  (LDS f32 atomics) ----
    for (int t = tid; t < EPG * CW; t += NTHREADS) {
        const int e = t / CW;
        const int f = t % CW;
        const unsigned p = sEdge[e];
        const int s = (int)(p & 0xFFFFu);
        const int d = (int)(p >> 16);
        const float w = sDis[s] * sDis[d];
        atomicAdd(&sAcc[d * HSTRIDE + f], sH[s * HSTRIDE + f] * w);
    }
    __syncthreads();

    // ---- ReLU -> global output (last layer) or next layer input ----
    if (LAST) {
        for (int t = tid; t < NPG * CW; t += NTHREADS) {
            const int i = t / CW;
            const int f = t % CW;
            const float v = sAcc[i * HSTRIDE + f];
            outg[i * 64 + c0 + f] = v > 0.f ? v : 0.f;
        }
        __syncthreads();   // before next pass reuses sH/sAcc
    } else {
        for (int t = tid; t < NPAD * FO; t += NTHREADS) {
            const int i = t / FO;
            const int f = t % FO;
            float v = (i < NPG) ? sAcc[i * HSTRIDE + f] : 0.f;
            sX[i * FO + f] = v > 0.f ? v : 0.f;   // zero-fills padding rows
        }
        __syncthreads();   // sX ready for next layer's GEMM
    }
}

__global__ __launch_bounds__(NTHREADS)
void gcn_fused_kernel(const float* __restrict__ x,
                      const int* __restrict__ esrc,
                      const int* __restrict__ edst,
                      const float* __restrict__ W1, const float* __restrict__ b1,
                      const float* __restrict__ W2, const float* __restrict__ b2,
                      const float* __restrict__ W3, const float* __restrict__ b3,
                      float* __restrict__ out)
{
    __shared__ float    sX[NPAD * 32];     // 20480 B
    __shared__ float    sH[NPAD * 32];     // 20480 B
    __shared__ float    sAcc[NPAD * 32];   // 20480 B
    __shared__ float    sDis[NPAD];        //   640 B
    __shared__ unsigned sEdge[EPG];        //  2368 B  -> total 64448 B

    const int g    = blockIdx.x;
    const int tid  = threadIdx.x;
    const int base = g * NPG;

    // degree init (self-loop = 1) + load x (fi=4) with zero padding rows
    for (int i = tid; i < NPAD; i += NTHREADS) sDis[i] = 1.0f;
    for (int t = tid; t < NPG * 4; t += NTHREADS) sX[t] = x[(size_t)base * 4 + t];
    for (int t = NPG * 4 + tid; t < NPAD * 4; t += NTHREADS) sX[t] = 0.f;
    __syncthreads();

    // load edges (localize to [0,148)) and accumulate in-degree
    for (int e = tid; e < EPG; e += NTHREADS) {
        const int s = esrc[(size_t)g * EPG + e] - base;
        const int d = edst[(size_t)g * EPG + e] - base;
        sEdge[e] = (unsigned)s | ((unsigned)d << 16);
        atomicAdd(&sDis[d], 1.0f);
    }
    __syncthreads();
    for (int i = tid; i < NPAD; i += NTHREADS) sDis[i] = rsqrtf(sDis[i]);
    __syncthreads();

    float* outg = out + (size_t)g * (NPG * 64);

    layer_pass< 4, 16, 16, false>(W1, b1, sX, sH, sAcc, sEdge, sDis, outg,  0, tid);
    layer_pass<16, 32, 32, false>(W2, b2, sX, sH, sAcc, sEdge, sDis, outg,  0, tid);
    layer_pass<32, 64, 32, true >(W3, b3, sX, sH, sAcc, sEdge, sDis, outg,  0, tid);
    layer_pass<32, 64, 32, true >(W3, b3, sX, sH, sAcc, sEdge, sDis, outg, 32, tid);
}

extern "C" void kernel_launch(void* const* d_in, const int* in_sizes, int n_in,
                              void* d_out, int out_size, void* d_ws, size_t ws_size,
                              hipStream_t stream)
{
    (void)in_sizes; (void)n_in; (void)out_size; (void)d_ws; (void)ws_size;

    const float* x    = (const float*)d_in[0];
    const int*   ei   = (const int*)  d_in[1];   // [2, E] stacked: src then dst
    // d_in[2] = batch (unused: graphs are block-constant by construction)
    const float* W1   = (const float*)d_in[3];
    const float* b1   = (const float*)d_in[4];
    const float* W2   = (const float*)d_in[5];
    const float* b2   = (const float*)d_in[6];
    const float* W3   = (const float*)d_in[7];
    const float* b3   = (const float*)d_in[8];
    float*       out  = (float*)d_out;

    const int E = B_GRAPHS * EPG;
    const int* esrc = ei;
    const int* edst = ei + E;

    gcn_fused_kernel<<<dim3(B_GRAPHS), dim3(NTHREADS), 0, stream>>>(
        x, esrc, edst, W1, b1, W2, b2, W3, b3, out);
}